// GraphConv_2259152797811
// MI455X (gfx1250) — compile-verified
//
#include <hip/hip_runtime.h>

typedef float v2f __attribute__((ext_vector_type(2)));
typedef float v8f __attribute__((ext_vector_type(8)));

#define D_IN  256
#define D_OUT 256

// ---------------------------------------------------------------------------
// Kernel 1: fused dual GEMM.  H0 = X @ W0  -> d_out,   H1 = X @ W1 -> ws
// grid = (ceil(N/128), D_OUT/16), block = 256 threads (8 wave32s).
// Each wave computes one 16x16 tile of H0 AND H1 (shared A fragment) using
// V_WMMA_F32_16X16X4_F32 over K=256 (64 WMMA steps per matrix).
// W0/W1 column slices (256x16 each) staged in LDS (32 KB).
// ---------------------------------------------------------------------------
__global__ __launch_bounds__(256) void graphconv_gemm2_wmma(
    const float* __restrict__ X,
    const float* __restrict__ W0,
    const float* __restrict__ W1,
    float* __restrict__ H0,      // -> d_out (pre-ReLU accumulation base)
    float* __restrict__ H1,      // -> workspace (messages source)
    int nrows)
{
    __shared__ float lw0[D_IN * 16];
    __shared__ float lw1[D_IN * 16];

    const int n0 = blockIdx.y * 16;

    // Cooperative load of the 16-column slices of both weights into LDS.
    // lw[k*16 + n] = W[k][n0+n]
    for (int i = threadIdx.x; i < D_IN * 16; i += 256) {
        const int k = i >> 4;
        const int n = i & 15;
        lw0[i] = W0[k * D_OUT + n0 + n];
        lw1[i] = W1[k * D_OUT + n0 + n];
    }
    __syncthreads();

    const int wave  = threadIdx.x >> 5;   // 0..7  (wave32)
    const int lane  = threadIdx.x & 31;
    const int row0  = blockIdx.x * 128 + wave * 16;

    // A-matrix 16x4 f32 layout (ISA 7.12.2):
    //   lanes 0-15 : M = lane,    VGPR0 = K+0, VGPR1 = K+1
    //   lanes 16-31: M = lane-16, VGPR0 = K+2, VGPR1 = K+3
    const int m     = lane & 15;
    const int khalf = (lane >> 4) << 1;   // 0 or 2

    const int  arow   = row0 + m;
    const bool arowok = arow < nrows;
    const float* xrow = X + (size_t)(arowok ? arow : 0) * D_IN;

    v8f acc0 = {};
    v8f acc1 = {};

    for (int kk = 0; kk < D_IN; kk += 4) {
        const int kb = kk + khalf;

        // A fragment: two consecutive K values -> one 8-byte load
        v2f a = *(const v2f*)(xrow + kb);

        // B fragments from LDS (mirrored layout: row K striped over lanes)
        v2f b0, b1;
        b0.x = lw0[(kb + 0) * 16 + m];
        b0.y = lw0[(kb + 1) * 16 + m];
        b1.x = lw1[(kb + 0) * 16 + m];
        b1.y = lw1[(kb + 1) * 16 + m];

        acc0 = __builtin_amdgcn_wmma_f32_16x16x4_f32(
                   false, a, false, b0, (short)0, acc0, false, false);
        acc1 = __builtin_amdgcn_wmma_f32_16x16x4_f32(
                   false, a, false, b1, (short)0, acc1, false, false);
    }

    // C/D 16x16 f32 layout: lanes 0-15 -> M = r, lanes 16-31 -> M = r + 8,
    // N = lane & 15, for VGPR r in 0..7.
    const int mbase = (lane >> 4) << 3;   // 0 or 8
    const int col   = n0 + (lane & 15);
    #pragma unroll
    for (int r = 0; r < 8; ++r) {
        const int rr = row0 + mbase + r;
        if (rr < nrows) {
            H0[(size_t)rr * D_OUT + col] = acc0[r];
            H1[(size_t)rr * D_OUT + col] = acc1[r];
        }
    }
}

// ---------------------------------------------------------------------------
// Kernel 2: edge scatter-add.  out[dst[e]] += H1[src[e]]  (per-element fp32
// atomics; coalesced 1KB row reads). One wave per edge, 8 floats per lane.
// ---------------------------------------------------------------------------
__global__ __launch_bounds__(256) void graphconv_scatter(
    const int* __restrict__ dst,
    const int* __restrict__ src,
    const float* __restrict__ H1,
    float* __restrict__ out,
    int nedges)
{
    const int gwave = (int)((blockIdx.x * (unsigned)blockDim.x + threadIdx.x) >> 5);
    const int lane  = threadIdx.x & 31;
    if (gwave >= nedges) return;

    const int d = dst[gwave];
    const int s = src[gwave];

    const float* srow = H1  + (size_t)s * D_OUT + lane * 8;
    float*       drow = out + (size_t)d * D_OUT + lane * 8;

    const float4 va = *(const float4*)(srow);
    const float4 vb = *(const float4*)(srow + 4);

    // Native global_atomic_add_f32 (no CAS loop)
    unsafeAtomicAdd(drow + 0, va.x);
    unsafeAtomicAdd(drow + 1, va.y);
    unsafeAtomicAdd(drow + 2, va.z);
    unsafeAtomicAdd(drow + 3, va.w);
    unsafeAtomicAdd(drow + 4, vb.x);
    unsafeAtomicAdd(drow + 5, vb.y);
    unsafeAtomicAdd(drow + 6, vb.z);
    unsafeAtomicAdd(drow + 7, vb.w);
}

// ---------------------------------------------------------------------------
// Kernel 3: in-place ReLU over d_out (float4 vectorized).
// ---------------------------------------------------------------------------
__global__ __launch_bounds__(256) void graphconv_relu(float4* __restrict__ out, int n4)
{
    const int i = blockIdx.x * blockDim.x + threadIdx.x;
    if (i < n4) {
        float4 v = out[i];
        v.x = fmaxf(v.x, 0.0f);
        v.y = fmaxf(v.y, 0.0f);
        v.z = fmaxf(v.z, 0.0f);
        v.w = fmaxf(v.w, 0.0f);
        out[i] = v;
    }
}

// ---------------------------------------------------------------------------
// Host launcher
// Inputs (setup_inputs order):
//   d_in[0] vertex_features : float32 [N, 256]
//   d_in[1] edge_index      : int32   [2, E]  (row 0 = dst, row 1 = src)
//   d_in[2] w0              : float32 [256, 256]
//   d_in[3] w1              : float32 [256, 256]
// d_out: float32 [N, 256]; d_ws used for H1 = X@W1 (N*256 floats).
// ---------------------------------------------------------------------------
extern "C" void kernel_launch(void* const* d_in, const int* in_sizes, int n_in,
                              void* d_out, int out_size, void* d_ws, size_t ws_size,
                              hipStream_t stream)
{
    const float* X    = (const float*)d_in[0];
    const int*   eidx = (const int*)d_in[1];
    const float* W0   = (const float*)d_in[2];
    const float* W1   = (const float*)d_in[3];
    float*       out  = (float*)d_out;
    float*       H1   = (float*)d_ws;

    const int nrows  = in_sizes[0] / D_IN;   // 50000
    const int nedges = in_sizes[1] / 2;      // 800000
    const int* dst = eidx;                   // row 0
    const int* src = eidx + nedges;          // row 1

    // 1) H0 -> d_out, H1 -> ws
    {
        dim3 grid((nrows + 127) / 128, D_OUT / 16);
        graphconv_gemm2_wmma<<<grid, 256, 0, stream>>>(X, W0, W1, out, H1, nrows);
    }

    // 2) out[dst] += H1[src]   (one wave per edge)
    {
        const long long total_threads = (long long)nedges * 32;
        const int blocks = (int)((total_threads + 255) / 256);
        graphconv_scatter<<<blocks, 256, 0, stream>>>(dst, src, H1, out, nedges);
    }

    // 3) ReLU in place
    {
        const int n4 = out_size / 4;
        graphconv_relu<<<(n4 + 255) / 256, 256, 0, stream>>>((float4*)out, n4);
    }
}